// Model_4449586118890
// MI455X (gfx1250) — compile-verified
//
#include <hip/hip_runtime.h>
#include <hip/hip_bf16.h>
#include <math.h>
#include <stdint.h>

// ---------------------------------------------------------------------------
// RPN proposal generation (top-k -> decode -> greedy NMS -> top-1000) for
// MI455X / gfx1250.  No matrix math exists in this op, so no WMMA; the
// CDNA5-specific path exercised is async global->LDS staging (ASYNCcnt),
// emitted via inline asm (the clang builtin's CUDA-device LangAS pointer
// params are not spellable from HIP source).
// ---------------------------------------------------------------------------

#define PRE_NMS   2000
#define CAP       2048            // candidate capacity (pow2 for bitonic sort)
#define POST_NMS  1000
#define IOU_T     0.7f
#define IMG_SZ    800.0f
#define BBOX_CLIP 4.1351665567423560f   // log(1000/16)

#if defined(__HIP_DEVICE_COMPILE__) && defined(__gfx1250__)
#define HAVE_ASYNC_LDS 1
// dst_lds: generic pointer into __shared__; low 32 bits of the flat address
// are the LDS byte offset (flat = aperture_base | as3_offset on AMDGPU).
// src_glb: generic pointer into global memory (64-bit VGPR pair, GV mode).
#define ASYNC_COPY_B128(dst_lds, src_glb)                                   \
  asm volatile("global_load_async_to_lds_b128 %0, %1, off"                  \
               :: "v"((unsigned)(uintptr_t)(dst_lds)),                      \
                  "v"((unsigned long long)(uintptr_t)(src_glb))             \
               : "memory")
#define ASYNC_WAIT0() asm volatile("s_wait_asynccnt 0x0" ::: "memory")
#else
#define HAVE_ASYNC_LDS 0
#define ASYNC_COPY_B128(dst_lds, src_glb) ((void)0)
#define ASYNC_WAIT0() ((void)0)
#endif

// Monotone map: float ordering -> unsigned ordering.
__device__ __forceinline__ uint32_t fkey(float f) {
  uint32_t u = __float_as_uint(f);
  return (u & 0x80000000u) ? ~u : (u | 0x80000000u);
}

// ---------------------------------------------------------------------------
// K0: zero counters, candidate buffer, output
// ---------------------------------------------------------------------------
__global__ void init_kernel(unsigned int* cnt, unsigned long long* cand,
                            float* out, int outN, int B) {
  int t = blockIdx.x * blockDim.x + threadIdx.x;
  int stride = gridDim.x * blockDim.x;
  for (int i = t; i < B; i += stride) cnt[i] = 0u;
  for (int i = t; i < B * CAP; i += stride) cand[i] = 0ull;
  for (int i = t; i < outN; i += stride) out[i] = 0.0f;
}

// ---------------------------------------------------------------------------
// K1: per-image 4-pass MSB radix select -> key of the 2000th-largest score
// ---------------------------------------------------------------------------
__global__ void radix_thresh_kernel(const float* __restrict__ scores, int A,
                                    unsigned int* __restrict__ thresh) {
  const int b = blockIdx.x;
  __shared__ unsigned int hist[256];
  __shared__ unsigned int sh_prefix, sh_mask, sh_rem;
  if (threadIdx.x == 0) { sh_prefix = 0u; sh_mask = 0u; sh_rem = PRE_NMS; }
  const float* src = scores + (size_t)b * A;

  for (int d = 0; d < 4; ++d) {
    const int shift = 24 - 8 * d;
    for (int v = threadIdx.x; v < 256; v += blockDim.x) hist[v] = 0u;
    __syncthreads();
    const unsigned int pm = sh_mask, pf = sh_prefix;
    for (int i = threadIdx.x; i < A; i += blockDim.x) {
      __builtin_prefetch(src + i + 4 * blockDim.x, 0, 0);
      uint32_t k = fkey(src[i]);
      if ((k & pm) == pf) atomicAdd(&hist[(k >> shift) & 255u], 1u);
    }
    __syncthreads();
    if (threadIdx.x == 0) {
      unsigned int rem = sh_rem;
      for (int v = 255; v >= 0; --v) {
        unsigned int c = hist[v];
        if (rem <= c) {
          sh_prefix = pf | ((unsigned int)v << shift);
          sh_mask   = pm | (0xFFu << shift);
          break;
        }
        rem -= c;
      }
      sh_rem = rem;
    }
    __syncthreads();
  }
  if (threadIdx.x == 0) thresh[b] = sh_prefix;
}

// ---------------------------------------------------------------------------
// K2: compact all keys >= T into candidate buffer as (key<<32)|~idx
// ---------------------------------------------------------------------------
__global__ void compact_kernel(const float* __restrict__ scores, int A,
                               const unsigned int* __restrict__ thresh,
                               unsigned int* __restrict__ cnt,
                               unsigned long long* __restrict__ cand) {
  const int b = blockIdx.y;
  const unsigned int T = thresh[b];
  const float* src = scores + (size_t)b * A;
  const int stride = gridDim.x * blockDim.x;
  for (int i = blockIdx.x * blockDim.x + threadIdx.x; i < A; i += stride) {
    uint32_t k = fkey(src[i]);
    if (k >= T) {
      unsigned int pos = atomicAdd(&cnt[b], 1u);
      if (pos < (unsigned int)CAP) {
        cand[(size_t)b * CAP + pos] =
            ((unsigned long long)k << 32) | (uint32_t)(~(uint32_t)i);
      }
    }
  }
}

// ---------------------------------------------------------------------------
// K3: bitonic sort 2048 u64 descending in LDS -> exact top-2000 order
//     (descending key, ascending index tiebreak via ~idx); gather logits.
// ---------------------------------------------------------------------------
__global__ void sort_select_kernel(const float* __restrict__ scores, int A,
                                   const unsigned long long* __restrict__ cand,
                                   unsigned int* __restrict__ sel_idx,
                                   float* __restrict__ sel_logit) {
  const int b = blockIdx.x;
  __shared__ unsigned long long s[CAP];
  for (int i = threadIdx.x; i < CAP; i += blockDim.x)
    s[i] = cand[(size_t)b * CAP + i];

  for (unsigned int k = 2; k <= CAP; k <<= 1) {
    for (unsigned int j = k >> 1; j > 0; j >>= 1) {
      __syncthreads();
      for (unsigned int idx = threadIdx.x; idx < CAP; idx += blockDim.x) {
        unsigned int ixj = idx ^ j;
        if (ixj > idx) {
          unsigned long long a = s[idx], c = s[ixj];
          bool desc = ((idx & k) == 0);
          if (desc ? (a < c) : (a > c)) { s[idx] = c; s[ixj] = a; }
        }
      }
    }
  }
  __syncthreads();
  for (int i = threadIdx.x; i < PRE_NMS; i += blockDim.x) {
    unsigned long long v = s[i];
    uint32_t idx = ~(uint32_t)v;
    if ((uint32_t)(v >> 32) == 0u) idx = 0u;  // unreachable pad guard
    sel_idx[b * PRE_NMS + i] = idx;
    sel_logit[b * PRE_NMS + i] = scores[(size_t)b * A + idx];
  }
}

// ---------------------------------------------------------------------------
// K4: decode the selected anchors+deltas -> clipped boxes (float4 per box)
// ---------------------------------------------------------------------------
__global__ void decode_kernel(const float* __restrict__ deltas,
                              const float* __restrict__ anchors,
                              const unsigned int* __restrict__ sel_idx,
                              float4* __restrict__ boxes, int A, int B) {
  int t = blockIdx.x * blockDim.x + threadIdx.x;
  if (t >= B * PRE_NMS) return;
  int b = t / PRE_NMS;
  unsigned int a = sel_idx[t];
  float4 an = ((const float4*)anchors)[a];
  float4 de = ((const float4*)deltas)[(size_t)b * A + a];

  float wa = an.z - an.x, ha = an.w - an.y;
  float cxa = an.x + 0.5f * wa, cya = an.y + 0.5f * ha;
  float dw = fminf(de.z, BBOX_CLIP), dh = fminf(de.w, BBOX_CLIP);
  float cx = de.x * wa + cxa, cy = de.y * ha + cya;
  float w = expf(dw) * wa, h = expf(dh) * ha;

  float4 o;
  o.x = fminf(fmaxf(cx - 0.5f * w, 0.0f), IMG_SZ);
  o.y = fminf(fmaxf(cy - 0.5f * h, 0.0f), IMG_SZ);
  o.z = fminf(fmaxf(cx + 0.5f * w, 0.0f), IMG_SZ);
  o.w = fminf(fmaxf(cy + 0.5f * h, 0.0f), IMG_SZ);
  boxes[t] = o;
}

// ---------------------------------------------------------------------------
// K5: per-image greedy NMS + rank scan + output.  Boxes staged into LDS via
//     CDNA5 async global->LDS (ASYNCcnt, inline asm).  Launch with exactly
//     1024 threads per block.
// ---------------------------------------------------------------------------
__global__ void nms_output_kernel(const float4* __restrict__ boxes,
                                  const float* __restrict__ sel_logit,
                                  float* __restrict__ out) {
  const int b = blockIdx.x;
  __shared__ float4 sbox[PRE_NMS];
  __shared__ unsigned int skeep[CAP];
  __shared__ unsigned int ssum[CAP];
  const float4* gb = boxes + (size_t)b * PRE_NMS;

#if HAVE_ASYNC_LDS
  for (int j = threadIdx.x; j < PRE_NMS; j += blockDim.x)
    ASYNC_COPY_B128(sbox + j, gb + j);
#else
  for (int j = threadIdx.x; j < PRE_NMS; j += blockDim.x) sbox[j] = gb[j];
#endif
  for (int j = threadIdx.x; j < CAP; j += blockDim.x)
    skeep[j] = (j < PRE_NMS) ? 1u : 0u;
#if HAVE_ASYNC_LDS
  ASYNC_WAIT0();
#endif
  __syncthreads();

  // Greedy NMS: sequential over score-descending order; parallel over j.
  for (int i = 0; i < PRE_NMS; ++i) {
    if (skeep[i]) {
      float4 bi = sbox[i];
      float ai = (bi.z - bi.x) * (bi.w - bi.y);
      for (int j = i + 1 + (int)threadIdx.x; j < PRE_NMS; j += blockDim.x) {
        if (skeep[j]) {
          float4 bj = sbox[j];
          float aj = (bj.z - bj.x) * (bj.w - bj.y);
          float lx = fmaxf(bi.x, bj.x), ly = fmaxf(bi.y, bj.y);
          float rx = fminf(bi.z, bj.z), ry = fminf(bi.w, bj.w);
          float iw = fmaxf(rx - lx, 0.0f), ih = fmaxf(ry - ly, 0.0f);
          float inter = iw * ih;
          float iou = inter / (ai + aj - inter + 1e-9f);
          if (iou > IOU_T) skeep[j] = 0u;
        }
      }
    }
    __syncthreads();
  }

  // Inclusive Hillis-Steele scan of keep flags (2048 elems, 1024 threads).
  {
    const int j0 = threadIdx.x, j1 = threadIdx.x + 1024;
    ssum[j0] = skeep[j0];
    ssum[j1] = skeep[j1];
    __syncthreads();
    for (int off = 1; off < CAP; off <<= 1) {
      unsigned int v0 = (j0 >= off) ? ssum[j0 - off] : 0u;
      unsigned int v1 = (j1 >= off) ? ssum[j1 - off] : 0u;
      __syncthreads();
      ssum[j0] += v0;
      ssum[j1] += v1;
      __syncthreads();
    }
  }

  // Emit first 1000 survivors as [x1,y1,x2,y2,sigmoid(logit)].
  for (int j = threadIdx.x; j < PRE_NMS; j += blockDim.x) {
    if (skeep[j]) {
      unsigned int r = ssum[j] - 1u;
      if (r < (unsigned int)POST_NMS) {
        float4 bx = sbox[j];
        float lg = sel_logit[b * PRE_NMS + j];
        float sc = 1.0f / (1.0f + expf(-lg));
        float* o = out + ((size_t)b * POST_NMS + r) * 5;
        o[0] = bx.x; o[1] = bx.y; o[2] = bx.z; o[3] = bx.w; o[4] = sc;
      }
    }
  }
}

// ---------------------------------------------------------------------------
// Host launcher
// ---------------------------------------------------------------------------
extern "C" void kernel_launch(void* const* d_in, const int* in_sizes, int n_in,
                              void* d_out, int out_size, void* d_ws,
                              size_t ws_size, hipStream_t stream) {
  const float* scores  = (const float*)d_in[0];  // (B, A)
  const float* deltas  = (const float*)d_in[1];  // (B, A, 4)
  const float* anchors = (const float*)d_in[2];  // (A, 4)
  const int A = in_sizes[2] / 4;
  const int B = in_sizes[0] / A;

  // Workspace layout (aligned, ~1.03 MB total for B=16).
  char* ws = (char*)d_ws;
  unsigned int* thresh = (unsigned int*)ws;                 // B u32
  unsigned int* cnt    = thresh + B;                        // B u32
  unsigned long long* cand = (unsigned long long*)(ws + 256);
  char* p = ws + 256 + (size_t)B * CAP * sizeof(unsigned long long);
  unsigned int* sel_idx = (unsigned int*)p; p += (size_t)B * PRE_NMS * 4;
  float* sel_logit = (float*)p;             p += (size_t)B * PRE_NMS * 4;
  float4* boxes = (float4*)p;

  init_kernel<<<256, 256, 0, stream>>>(cnt, cand, (float*)d_out, out_size, B);
  radix_thresh_kernel<<<B, 1024, 0, stream>>>(scores, A, thresh);
  compact_kernel<<<dim3(48, B), 256, 0, stream>>>(scores, A, thresh, cnt, cand);
  sort_select_kernel<<<B, 1024, 0, stream>>>(scores, A, cand, sel_idx, sel_logit);
  decode_kernel<<<(B * PRE_NMS + 255) / 256, 256, 0, stream>>>(
      deltas, anchors, sel_idx, boxes, A, B);
  nms_output_kernel<<<B, 1024, 0, stream>>>(boxes, sel_logit, (float*)d_out);
}